// VirtualNodePyg_78700980731921
// MI455X (gfx1250) — compile-verified
//
#include <hip/hip_runtime.h>

// ---------------------------------------------------------------------------
// VirtualNodePyg on MI455X (gfx1250, wave32).
// Phase 1: deterministic segment-sum (sorted batch -> per-graph binary search),
//          NT loads of streaming x.
// Phase 2: [B,128]x[128,128] f32 GEMM via V_WMMA_F32_16X16X4_F32 + bias/relu/res
// Phase 3: gather-broadcast add; NT load of x, NT store of x_out, cached vn_out.
// ---------------------------------------------------------------------------

typedef __attribute__((ext_vector_type(2))) float v2f;
typedef __attribute__((ext_vector_type(4))) float v4f;
typedef __attribute__((ext_vector_type(8))) float v8f;

#define FEAT 128

__device__ __forceinline__ long long lower_bound_i64(const long long* __restrict__ a,
                                                     long long n, long long key) {
  long long lo = 0, hi = n;
  while (lo < hi) {
    long long mid = (lo + hi) >> 1;
    if (a[mid] < key) lo = mid + 1; else hi = mid;
  }
  return lo;
}

// One block per graph; 128 threads = one feature column each.
// apool[b][d] = vn_x[b][d] + sum_{i in segment b} x[i][d]
// x is streamed exactly once here -> non-temporal loads keep L2 for hot data.
__global__ void vnp_pool_kernel(const float* __restrict__ x,
                                const float* __restrict__ vn_x,
                                const long long* __restrict__ batch,
                                float* __restrict__ apool,
                                long long N) {
  const int b = blockIdx.x;
  __shared__ long long s_range[2];
  if (threadIdx.x < 2) {
    s_range[threadIdx.x] = lower_bound_i64(batch, N, (long long)b + (long long)threadIdx.x);
  }
  __syncthreads();
  const long long start = s_range[0];
  const long long end   = s_range[1];
  const int d = threadIdx.x;

  float s0 = 0.f, s1 = 0.f, s2 = 0.f, s3 = 0.f;
  long long i = start;
  for (; i + 4 <= end; i += 4) {
    s0 += __builtin_nontemporal_load(&x[(i + 0) * FEAT + d]);
    s1 += __builtin_nontemporal_load(&x[(i + 1) * FEAT + d]);
    s2 += __builtin_nontemporal_load(&x[(i + 2) * FEAT + d]);
    s3 += __builtin_nontemporal_load(&x[(i + 3) * FEAT + d]);
  }
  for (; i < end; ++i) s0 += __builtin_nontemporal_load(&x[i * FEAT + d]);

  const long long o = (long long)b * FEAT + d;
  apool[o] = vn_x[o] + ((s0 + s1) + (s2 + s3));
}

// vn_out = vn_x + relu(apool @ W + bias)
// Grid: B/16 blocks x 256 threads (8 waves). Wave w computes the 16x16 tile at
// (mtile*16, w*16) with 32 chained V_WMMA_F32_16X16X4_F32 ops over K=128.
//
// VGPR layouts per CDNA5 ISA 7.12.2:
//   A (16x4 f32): lanes 0-15 -> M=lane, v0=K+0, v1=K+1; lanes 16-31 -> v0=K+2, v1=K+3
//   B (4x16 f32): row-striped across lanes: lanes 0-15 -> N=lane, v0=row K+0, v1=row K+1;
//                 lanes 16-31 -> v0=row K+2, v1=row K+3
//   C/D (16x16 f32): vgpr r: lanes 0-15 -> (M=mbase+r, N=lane); lanes 16-31 -> (M=mbase+8+r)
__global__ void vnp_gemm_kernel(const float* __restrict__ apool,
                                const float* __restrict__ vn_x,
                                const float* __restrict__ W,
                                const float* __restrict__ bias,
                                float* __restrict__ vn_out) {
  const int wave   = threadIdx.x >> 5;
  const int lane   = threadIdx.x & 31;
  const int lhalf  = lane >> 4;       // 0: lanes 0-15, 1: lanes 16-31
  const int l15    = lane & 15;
  const int mbase  = blockIdx.x * 16;
  const int nbase  = wave * 16;

  v8f acc = {};
  const int arow = mbase + l15;       // A row owned by this lane
  const int ncol = nbase + l15;       // B/C column owned by this lane

  for (int k = 0; k < FEAT; k += 4) {
    const int ka = k + lhalf * 2;
    v2f afrag;
    afrag.x = apool[(long long)arow * FEAT + ka + 0];
    afrag.y = apool[(long long)arow * FEAT + ka + 1];
    v2f bfrag;
    bfrag.x = W[(long long)(ka + 0) * FEAT + ncol];
    bfrag.y = W[(long long)(ka + 1) * FEAT + ncol];
    acc = __builtin_amdgcn_wmma_f32_16x16x4_f32(
        /*neg_a=*/false, afrag, /*neg_b=*/false, bfrag,
        /*c_mod=*/(short)0, acc, /*reuse_a=*/false, /*reuse_b=*/false);
  }

  const float bv = bias[ncol];
#pragma unroll
  for (int r = 0; r < 8; ++r) {
    const int row = mbase + lhalf * 8 + r;
    float v = acc[r] + bv;
    v = v > 0.f ? v : 0.f;                       // relu
    const long long o = (long long)row * FEAT + ncol;
    vn_out[o] = vn_x[o] + v;                     // residual on virtual node
  }
}

// x_out[i][:] = x[i][:] + vn_out[batch[i]][:], float4-vectorized.
// x / x_out are streaming (touched once) -> NT load / NT store.
// vn_out rows are reused ~N/B times -> default (cached) loads, stays in L2.
__global__ void vnp_broadcast_kernel(const float* __restrict__ x,
                                     const long long* __restrict__ batch,
                                     const float* __restrict__ vn_out,
                                     float* __restrict__ x_out,
                                     long long units) {
  long long u = (long long)blockIdx.x * blockDim.x + threadIdx.x;
  const long long stride = (long long)gridDim.x * blockDim.x;
  for (; u < units; u += stride) {
    const long long row = u >> 5;                // 32 float4 per 128-wide row
    const int       q   = (int)(u & 31);
    const long long g   = batch[row];
    const v4f xv = __builtin_nontemporal_load((const v4f*)(x + row * FEAT + q * 4));
    const v4f vv = *(const v4f*)(vn_out + g * FEAT + q * 4);
    v4f ov = xv + vv;
    __builtin_nontemporal_store(ov, (v4f*)(x_out + row * FEAT + q * 4));
  }
}

extern "C" void kernel_launch(void* const* d_in, const int* in_sizes, int n_in,
                              void* d_out, int out_size, void* d_ws, size_t ws_size,
                              hipStream_t stream) {
  const float*     x     = (const float*)d_in[0];
  const float*     vn_x  = (const float*)d_in[1];
  const long long* batch = (const long long*)d_in[2];   // int64 in reference
  const float*     W     = (const float*)d_in[3];
  const float*     bias  = (const float*)d_in[4];

  const long long N = (long long)in_sizes[0] / FEAT;    // 2,000,000
  const int       B = in_sizes[1] / FEAT;               // 16384

  float* x_out  = (float*)d_out;                        // [N,128]
  float* vn_out = x_out + N * (long long)FEAT;          // [B,128] (second tuple elem)
  float* apool  = (float*)d_ws;                         // [B,128] = vn_x + segment_sum

  // Phase 1: deterministic per-graph segment sum (+ vn_x) into workspace.
  vnp_pool_kernel<<<B, FEAT, 0, stream>>>(x, vn_x, batch, apool, N);

  // Phase 2: WMMA f32 GEMM + bias + relu + residual -> vn_out.
  vnp_gemm_kernel<<<B / 16, 256, 0, stream>>>(apool, vn_x, W, bias, vn_out);

  // Phase 3: gather-broadcast add back to every node.
  const long long units  = N * (FEAT / 4);
  const int       thr    = 256;
  long long blocks = (units + thr - 1) / thr;
  if (blocks > 262144) blocks = 262144;                 // grid-stride covers rest
  vnp_broadcast_kernel<<<(unsigned)blocks, thr, 0, stream>>>(x, batch, vn_out, x_out, units);
}